// Pred3DPose_76476187673130
// MI455X (gfx1250) — compile-verified
//
#include <hip/hip_runtime.h>

typedef __attribute__((ext_vector_type(2))) float v2f;
typedef __attribute__((ext_vector_type(8))) float v8f;

// ---------------------------------------------------------------------------
// Generic fused MLP block: out = ReLU(LN(A @ W1 + b1)) @ W2 + b2
// One wave (32 lanes) per 16-row tile.  fp32 WMMA 16x16x4, K looped by 4 from
// zero-padded LDS tiles (branch-free inner loop, EXEC stays all-ones).
//
// A is a *view*:  A(r, k) = base[(r/G)*GS + (r%G)*RS + k*CS]
//   contiguous matrix:  G = M, GS = 0, RS = row_stride, CS = 1
//   swapped-last-two:   G = rows_per_group, GS = group_stride, RS = 1, CS = C
// W1: [K x H], W2: [H x N] row-major.  out row stride oStride, col off oOff.
// ---------------------------------------------------------------------------
__global__ __launch_bounds__(32) void mlp16_wmma(
    const float* __restrict__ A, int M, int K,
    int G, int GS, int RS, int CS,
    const float* __restrict__ W1, const float* __restrict__ b1,
    const float* __restrict__ g,  const float* __restrict__ Beta,
    const float* __restrict__ W2, const float* __restrict__ b2,
    int H, int N,
    float* __restrict__ out, int oStride, int oOff)
{
    __shared__ float Asm[16][304];   // A tile, zero-padded to Kpad (K <= 300)
    __shared__ float Bsm[16][304];   // B tile transposed: Bsm[n][k]
    __shared__ float hbuf[16][112];  // hidden tile (H <= 100)

    const int lane = threadIdx.x;    // 0..31, wave32
    const int m16  = lane & 15;
    const int half = lane >> 4;      // 0 or 1
    const int row0 = blockIdx.x * 16;

    // ---- stage A tile into LDS (gather through the view, zero-pad) --------
    const int Kpad = (K + 3) & ~3;
    for (int idx = lane; idx < 16 * Kpad; idx += 32) {
        int r  = idx / Kpad;
        int c  = idx - r * Kpad;
        int gr = row0 + r;
        float v = 0.f;
        if (gr < M && c < K)
            v = A[(size_t)(gr / G) * GS + (size_t)(gr % G) * RS + (size_t)c * CS];
        Asm[r][c] = v;
    }
    __syncthreads();

    // ---------------- GEMM1: hidden = A_tile(16xK) @ W1(KxH) ----------------
    const int hTiles = (H + 15) >> 4;
    for (int nt = 0; nt < hTiles; ++nt) {
        const int n0  = nt << 4;
        const int col = n0 + m16;
        // stage W1 tile (transposed, zero-padded)
        for (int idx = lane; idx < 16 * Kpad; idx += 32) {
            int k = idx >> 4;
            int j = idx & 15;
            float v = 0.f;
            if (k < K && (n0 + j) < H) v = W1[(size_t)k * H + n0 + j];
            Bsm[j][k] = v;
        }
        __syncthreads();

        v8f acc = {};
        for (int k0 = 0; k0 < Kpad; k0 += 4) {
            const int kk = k0 + 2 * half;       // fragment K split by lane half
            v2f a, b;
            a.x = Asm[m16][kk];  a.y = Asm[m16][kk + 1];
            b.x = Bsm[m16][kk];  b.y = Bsm[m16][kk + 1];
            acc = __builtin_amdgcn_wmma_f32_16x16x4_f32(
                      false, a, false, b, (short)0, acc, false, false);
        }
        __syncthreads();

        if (col < H) {
            const float bias = b1[col];
            #pragma unroll
            for (int v = 0; v < 8; ++v)
                hbuf[v + 8 * half][col] = acc[v] + bias;   // C/D row = v + 8*half
        }
        __syncthreads();
    }

    // ---------------- LayerNorm + ReLU, one row per lane (0..15) ------------
    const int Hpad = (H + 3) & ~3;
    if (lane < 16) {
        float mu = 0.f;
        for (int c = 0; c < H; ++c) mu += hbuf[lane][c];
        mu /= (float)H;
        float var = 0.f;
        for (int c = 0; c < H; ++c) { float d = hbuf[lane][c] - mu; var += d * d; }
        var /= (float)H;
        const float inv = rsqrtf(var + 1e-5f);
        for (int c = 0; c < H; ++c) {
            float v = (hbuf[lane][c] - mu) * inv * g[c] + Beta[c];
            hbuf[lane][c] = v > 0.f ? v : 0.f;
        }
        for (int c = H; c < Hpad; ++c) hbuf[lane][c] = 0.f;  // zero-pad for GEMM2
    }
    __syncthreads();

    // ---------------- GEMM2: out = hidden(16xH) @ W2(HxN) + b2 -------------
    const int nTiles = (N + 15) >> 4;
    for (int nt = 0; nt < nTiles; ++nt) {
        const int n0  = nt << 4;
        const int col = n0 + m16;
        for (int idx = lane; idx < 16 * Hpad; idx += 32) {
            int k = idx >> 4;
            int j = idx & 15;
            float v = 0.f;
            if (k < H && (n0 + j) < N) v = W2[(size_t)k * N + n0 + j];
            Bsm[j][k] = v;
        }
        __syncthreads();

        v8f acc = {};
        for (int k0 = 0; k0 < Hpad; k0 += 4) {
            const int kk = k0 + 2 * half;
            v2f a, b;
            a.x = hbuf[m16][kk]; a.y = hbuf[m16][kk + 1];
            b.x = Bsm[m16][kk];  b.y = Bsm[m16][kk + 1];
            acc = __builtin_amdgcn_wmma_f32_16x16x4_f32(
                      false, a, false, b, (short)0, acc, false, false);
        }
        __syncthreads();

        if (col < N) {
            const float bias = b2[col];
            #pragma unroll
            for (int v = 0; v < 8; ++v) {
                const int grow = row0 + v + 8 * half;
                if (grow < M)
                    out[(size_t)grow * oStride + oOff + col] = acc[v] + bias;
            }
        }
        __syncthreads();
    }
}

// ---------------------------------------------------------------------------
// Elementwise helpers (operate on the swapped view of Sout = [14,100,17],
// whose logical "h" view is h(f, r, c) = Sout[f*1700 + c*17 + r])
// ---------------------------------------------------------------------------
__global__ void gather_sub_k(const float* __restrict__ S, int f0a, int f0b,
                             float* __restrict__ out)
{
    int idx = blockIdx.x * blockDim.x + threadIdx.x;   // over 170*100
    if (idx >= 17000) return;
    int gr = idx / 100;
    int k  = idx - gr * 100;
    int f  = gr / 17;
    int r  = gr - f * 17;
    int o  = k * 17 + r;
    out[idx] = S[(f + f0a) * 1700 + o] - S[(f + f0b) * 1700 + o];
}

__global__ void gather_copy_k(const float* __restrict__ S, int f0,
                              float* __restrict__ out, int outStride, int outOff)
{
    int idx = blockIdx.x * blockDim.x + threadIdx.x;   // over 170*100
    if (idx >= 17000) return;
    int gr = idx / 100;
    int k  = idx - gr * 100;
    int f  = gr / 17;
    int r  = gr - f * 17;
    out[(size_t)gr * outStride + outOff + k] = S[(f + f0) * 1700 + k * 17 + r];
}

__global__ void zero_k(float* __restrict__ p, int n)
{
    int idx = blockIdx.x * blockDim.x + threadIdx.x;
    if (idx < n) p[idx] = 0.f;
}

// ---------------------------------------------------------------------------
// Host orchestration
// ---------------------------------------------------------------------------
extern "C" void kernel_launch(void* const* d_in, const int* in_sizes, int n_in,
                              void* d_out, int out_size, void* d_ws, size_t ws_size,
                              hipStream_t stream)
{
    (void)in_sizes; (void)n_in; (void)out_size; (void)ws_size;

    const float* x = (const float*)d_in[0];
    auto Wp = [&](int i) { return (const float*)d_in[i]; };
    // weight base indices: i=1, c=7, s=13, m=19, n=25, t=31, k=37, p=43
    const int Bi = 1, Bc = 7, Bs = 13, Bm = 19, Bn = 25, Bt = 31, Bk = 37, Bp = 43;

    float* ws   = (float*)d_ws;
    float* Y1   = ws;              // [14,17,10,100] 238000
    float* Cout = Y1   + 238000;   // [14,17,100]     23800
    float* Sout = Cout + 23800;    // [14,100,17]     23800
    float* Da   = Sout + 23800;    // [170,100]       17000
    float* Db   = Da   + 17000;    // [170,100]       17000
    float* CatA = Db   + 17000;    // [170,200]       34000
    float* CatB = CatA + 34000;    // [170,200]       34000
    float* Cat3 = CatB + 34000;    // [170,300]       51000
    float* Fin  = Cat3 + 51000;    // [204,100]       20400

    auto mlp = [&](const float* Ain, int M, int K, int G, int GS, int RS, int CS,
                   int base, int H, int N, float* o, int oS, int oOff) {
        mlp16_wmma<<<(M + 15) / 16, 32, 0, stream>>>(
            Ain, M, K, G, GS, RS, CS,
            Wp(base), Wp(base + 1), Wp(base + 2), Wp(base + 3),
            Wp(base + 4), Wp(base + 5),
            H, N, o, oS, oOff);
    };
    auto ew = [&](int n) { return dim3((n + 255) / 256); };

    // 1) mlp_i: x [2380,3] -> Y1 [2380,100]  (= [14,17,10,100])
    mlp(x, 2380, 3, /*G*/2380, 0, 3, 1, Bi, 20, 100, Y1, 100, 0);
    // 2) mlp_c on swapped view of Y1 ([14,17,100,10]): rows 23800, K=10
    //    A(r,k) = Y1[(r/100)*1000 + (r%100)*1 + k*100]
    mlp(Y1, 23800, 10, /*G*/100, 1000, 1, 100, Bc, 8, 1, Cout, 1, 0);
    // 3) mlp_s on swapped view of Cout ([14,100,17]): rows 1400, K=17
    //    A(r,k) = Cout[(r/100)*1700 + (r%100)*1 + k*100]
    mlp(Cout, 1400, 17, /*G*/100, 1700, 1, 100, Bs, 34, 17, Sout, 17, 0);

    // h view: h(f,r,c) = Sout[f*1700 + c*17 + r]   ([14,17,100])
    // view params for h[f0:f0+10] as [170,100]: G=17, GS=1700, RS=1, CS=17
    // 4) Da = h[1:11] - h[0:10]
    gather_sub_k<<<ew(17000), 256, 0, stream>>>(Sout, 1, 0, Da);
    // 5) mf = mlp_m(Da)        -> CatA[:,100:200]
    mlp(Da, 170, 100, 170, 0, 100, 1, Bm, 20, 100, CatA, 200, 100);
    // 6) pf_n = mlp_n(h[1:11]) -> CatA[:,0:100]
    mlp(Sout + 1700, 170, 100, 17, 1700, 1, 17, Bn, 20, 100, CatA, 200, 0);
    // 7) pf = mlp_t(CatA)      -> Cat3[:,0:100]
    mlp(CatA, 170, 200, 170, 0, 200, 1, Bt, 20, 100, Cat3, 300, 0);

    // 8) Db = h[4:14] - h[3:13]
    gather_sub_k<<<ew(17000), 256, 0, stream>>>(Sout, 4, 3, Db);
    // 9) mb = mlp_m(Db)        -> CatB[:,100:200]
    mlp(Db, 170, 100, 170, 0, 100, 1, Bm, 20, 100, CatB, 200, 100);
    // 10) nf_n = mlp_n(h[3:13]) -> CatB[:,0:100]
    mlp(Sout + 3 * 1700, 170, 100, 17, 1700, 1, 17, Bn, 20, 100, CatB, 200, 0);
    // 11) nf = mlp_t(CatB)     -> Cat3[:,200:300]
    mlp(CatB, 170, 200, 170, 0, 200, 1, Bt, 20, 100, Cat3, 300, 200);

    // 12) h[2:12] -> Cat3[:,100:200]
    gather_copy_k<<<ew(17000), 256, 0, stream>>>(Sout, 2, Cat3, 300, 100);
    // 13) comb = mlp_k(Cat3)   -> Fin rows 0..169
    mlp(Cat3, 170, 300, 170, 0, 300, 1, Bk, 100, 100, Fin, 100, 0);
    // 14) zero rows 170..203 of Fin (frames 10,11 stay zero)
    zero_k<<<ew(3400), 256, 0, stream>>>(Fin + 17000, 3400);
    // 15) final = mlp_p(Fin)   -> d_out [204,3]
    mlp(Fin, 204, 100, 204, 0, 100, 1, Bp, 30, 3, (float*)d_out, 3, 0);
}